// TransformerSeqLayer_29927332118891
// MI455X (gfx1250) — compile-verified
//
#include <hip/hip_runtime.h>
#include <hip/hip_bf16.h>

typedef __attribute__((ext_vector_type(2))) float v2f;
typedef __attribute__((ext_vector_type(8))) float v8f;

// ---------------------------------------------------------------------------
// CDNA5 fp32 WMMA: D(16x16,f32) = A(16x4,f32) x B(4x16,f32) + C
// A frag (per lane): lanes 0-15 rows M=0..15 {K=0,K=1}; lanes 16-31 {K=2,K=3}
// B frag (per lane): lanes 0-15 cols N=0..15 {K=0,K=1}; lanes 16-31 {K=2,K=3}
// C/D: VGPR r -> M = r + 8*(lane/16), N = lane%16
// ---------------------------------------------------------------------------
__device__ __forceinline__ v8f wmma4(v2f a, v2f b, v8f c) {
  return __builtin_amdgcn_wmma_f32_16x16x4_f32(false, a, false, b, (short)0, c,
                                               false, false);
}

__device__ __forceinline__ v8f v8zero() {
  v8f z;
#pragma unroll
  for (int i = 0; i < 8; ++i) z[i] = 0.0f;
  return z;
}

// ---------------------------------------------------------------------------
// Shared software-pipelined 64x64-tile GEMM core. 128 threads / 4 waves,
// wave w -> rows [w*16, w*16+16), all 64 cols. KC = 32.
// LDS: As row-major stride 36 (16B-aligned stores, aligned b64 frag loads),
//      Bs2 k-pair interleaved: Bs2[p*128 + n*2 + j] = B[2p+j, n]
//      -> every A/B fragment is one aligned ds_load_b64, zero repack movs.
// Next chunk's global loads issued before the WMMA burst (latency overlap).
// ---------------------------------------------------------------------------
template <typename AL, typename BL>
__device__ __forceinline__ void gemm_tiles(int K, AL aload, BL bload,
                                           float* As, float* Bs2, v8f* acc,
                                           int tid) {
  const int wave = tid >> 5;
  const int lane = tid & 31;
  const int ln = lane & 15;
  const int hi = lane >> 4;
  float4 ra[4], rb[4];
#pragma unroll
  for (int i = 0; i < 4; ++i) {                       // A chunk: 64x32
    int idx = tid + i * 128;
    ra[i] = aload(idx >> 3, (idx & 7) << 2, 0);
  }
#pragma unroll
  for (int i = 0; i < 2; ++i) {                       // B chunk: 32x64
    int u = tid + i * 128;
    int p = u >> 4, g = (u & 15) << 2;
    rb[2 * i] = bload(2 * p, g, 0);
    rb[2 * i + 1] = bload(2 * p + 1, g, 0);
  }
  for (int k0 = 0; k0 < K; k0 += 32) {
    __syncthreads();                                  // prev compute done
#pragma unroll
    for (int i = 0; i < 4; ++i) {
      int idx = tid + i * 128;
      *(float4*)&As[(idx >> 3) * 36 + ((idx & 7) << 2)] = ra[i];
    }
#pragma unroll
    for (int i = 0; i < 2; ++i) {
      int u = tid + i * 128;
      int p = u >> 4, g = (u & 15) << 2;
      float4 lo, hv;
      lo.x = rb[2 * i].x; lo.y = rb[2 * i + 1].x;
      lo.z = rb[2 * i].y; lo.w = rb[2 * i + 1].y;
      hv.x = rb[2 * i].z; hv.y = rb[2 * i + 1].z;
      hv.z = rb[2 * i].w; hv.w = rb[2 * i + 1].w;
      *(float4*)&Bs2[p * 128 + g * 2] = lo;
      *(float4*)&Bs2[p * 128 + g * 2 + 4] = hv;
    }
    __syncthreads();
    int kn = k0 + 32;
    if (kn < K) {                                     // prefetch next chunk
#pragma unroll
      for (int i = 0; i < 4; ++i) {
        int idx = tid + i * 128;
        ra[i] = aload(idx >> 3, (idx & 7) << 2, kn);
      }
#pragma unroll
      for (int i = 0; i < 2; ++i) {
        int u = tid + i * 128;
        int p = u >> 4, g = (u & 15) << 2;
        rb[2 * i] = bload(2 * p, g, kn);
        rb[2 * i + 1] = bload(2 * p + 1, g, kn);
      }
    }
    const int arow = (wave * 16 + ln) * 36;
#pragma unroll
    for (int kk = 0; kk < 32; kk += 4) {
      v2f a = *(const v2f*)&As[arow + kk + 2 * hi];
      const float* bb = &Bs2[((kk >> 1) + hi) * 128 + ln * 2];
#pragma unroll
      for (int t = 0; t < 4; ++t) {
        v2f bv = *(const v2f*)(bb + t * 32);
        acc[t] = wmma4(a, bv, acc[t]);
      }
    }
  }
}

// ---------------------------------------------------------------------------
// Generic GEMM: C[M,N] = A[M,K] @ B[K,N] (row-major), M%64==0,N%64==0,K%32==0
// headsStoreT!=0: store remap (row=b*T+t, col=kh*128+d)->[(b*8+kh)*T+t, d]
// headsLoadT !=0: A load remap (same mapping, merges heads back)
// ---------------------------------------------------------------------------
__global__ __launch_bounds__(128)
void gemm_f32_wmma(const float* __restrict__ A, const float* __restrict__ B,
                   float* __restrict__ C, int M, int N, int K,
                   int lda, int ldb, int ldc,
                   long long sA, long long sB, long long sC,
                   int headsStoreT, int headsLoadT) {
  const float* Ab = A + (long long)blockIdx.z * sA;
  const float* Bb = B + (long long)blockIdx.z * sB;
  float*       Cb = C + (long long)blockIdx.z * sC;
  const int m_blk = blockIdx.y * 64;
  const int n_blk = blockIdx.x * 64;
  const int tid = threadIdx.x;
  const int wave = tid >> 5;
  const int lane = tid & 31;
  const int ln = lane & 15;
  const int hi = lane >> 4;

  __shared__ float As[64 * 36];
  __shared__ float Bs2[16 * 128];
  v8f acc[4];
#pragma unroll
  for (int t = 0; t < 4; ++t) acc[t] = v8zero();

  auto aload = [&](int r, int c4, int k0) -> float4 {
    int gm = m_blk + r;
    int gk = k0 + c4;
    const float* src;
    if (headsLoadT) {
      int bb = gm / headsLoadT;
      int tt = gm - bb * headsLoadT;
      int kh = gk >> 7;
      int d = gk & 127;
      src = Ab + (((long long)(bb * 8 + kh)) * headsLoadT + tt) * 128 + d;
    } else {
      src = Ab + (long long)gm * lda + gk;
    }
    return *(const float4*)src;
  };
  auto bload = [&](int r, int c4, int k0) -> float4 {
    return *(const float4*)(Bb + (long long)(k0 + r) * ldb + n_blk + c4);
  };
  gemm_tiles(K, aload, bload, As, Bs2, acc, tid);

#pragma unroll
  for (int t = 0; t < 4; ++t) {
    int n = n_blk + t * 16 + ln;
#pragma unroll
    for (int r = 0; r < 8; ++r) {
      int m = m_blk + wave * 16 + 8 * hi + r;
      float val = acc[t][r];
      if (headsStoreT) {
        int bb = m / headsStoreT;
        int tt = m - bb * headsStoreT;
        int kh = n >> 7;
        int d = n & 127;
        Cb[(((long long)(bb * 8 + kh)) * headsStoreT + tt) * 128 + d] = val;
      } else {
        Cb[(long long)m * ldc + n] = val;
      }
    }
  }
}

// ---------------------------------------------------------------------------
// Banded QK^T:  band[h][m][l] += q_h[m,:] . k_h[m+l,:]   (l = n-m, 0<=l<512)
// grid(16, 32, 32heads), 4 waves/block; wave = one 16x16 (m,n) tile.
// ---------------------------------------------------------------------------
__global__ __launch_bounds__(128)
void attn_band_cont(const float* __restrict__ q, const float* __restrict__ k,
                    float* __restrict__ band) {
  const int hb = blockIdx.z;
  const int m0 = blockIdx.y * 16;
  const int wave = threadIdx.x >> 5;
  const int lane = threadIdx.x & 31;
  const int n0 = (blockIdx.x * 4 + wave) * 16;
  if (n0 + 15 < m0 || n0 > m0 + 526) return;   // wave-uniform skip
  const int ln = lane & 15;
  const int hi = lane >> 4;
  const float* qrow = q + (size_t)hb * 512 * 128 + (m0 + ln) * 128 + 2 * hi;
  const float* krow = k + (size_t)hb * 1024 * 128 + (n0 + ln) * 128 + 2 * hi;
  v8f acc = v8zero();
#pragma unroll
  for (int kk = 0; kk < 128; kk += 4) {
    v2f a = *(const v2f*)(qrow + kk);
    v2f b = *(const v2f*)(krow + kk);
    acc = wmma4(a, b, acc);
  }
  float* bnd = band + (size_t)hb * 512 * 512;
  const int n = n0 + ln;
#pragma unroll
  for (int r = 0; r < 8; ++r) {
    int m = m0 + 8 * hi + r;
    int l = n - m;
    if (l >= 0 && l < 512) bnd[m * 512 + l] += acc[r];
  }
}

// ---------------------------------------------------------------------------
// Row softmax with 1/sqrt(128) scaling over band rows (32*512 rows x 512)
// ---------------------------------------------------------------------------
__global__ __launch_bounds__(256)
void softmax_band(float* __restrict__ band) {
  const float scale = 0.088388347648318447f;   // 1/sqrt(128)
  float* p = band + (size_t)blockIdx.x * 512;
  const int tid = threadIdx.x;
  __shared__ float sr[256];
  float v0 = p[tid] * scale;
  float v1 = p[tid + 256] * scale;
  sr[tid] = fmaxf(v0, v1);
  __syncthreads();
  for (int s = 128; s > 0; s >>= 1) {
    if (tid < s) sr[tid] = fmaxf(sr[tid], sr[tid + s]);
    __syncthreads();
  }
  float mx = sr[0];
  __syncthreads();
  float e0 = expf(v0 - mx), e1 = expf(v1 - mx);
  sr[tid] = e0 + e1;
  __syncthreads();
  for (int s = 128; s > 0; s >>= 1) {
    if (tid < s) sr[tid] += sr[tid + s];
    __syncthreads();
  }
  float inv = 1.0f / sr[0];
  p[tid] = e0 * inv;
  p[tid + 256] = e1 * inv;
}

// ---------------------------------------------------------------------------
// out_h[m,d] = sum_n skewP[m,n]*v_h[n,d], skewP[m,n] = p[m,n-m] if 0<=n-m<512
// A fragment gathered with shifted/masked loads. grid(2, 32, 32heads).
// ---------------------------------------------------------------------------
__global__ __launch_bounds__(128)
void attn_out_kernel(const float* __restrict__ p, const float* __restrict__ v,
                     float* __restrict__ outh) {
  const int hb = blockIdx.z;
  const int m0 = blockIdx.y * 16;
  const int wave = threadIdx.x >> 5;
  const int lane = threadIdx.x & 31;
  const int d0 = (blockIdx.x * 4 + wave) * 16;
  const int ln = lane & 15;
  const int hi = lane >> 4;
  const int arow = m0 + ln;
  const float* prow = p + (size_t)hb * 512 * 512 + (size_t)arow * 512;
  const float* vh = v + (size_t)hb * 1024 * 128;
  v8f acc = v8zero();
#pragma unroll 4
  for (int n = m0; n < m0 + 528; n += 4) {
    int kk = n + 2 * hi;
    int c = kk - arow;
    v2f a;
    a.x = (c >= 0 && c < 512) ? prow[c] : 0.0f;
    a.y = (c + 1 >= 0 && c + 1 < 512) ? prow[c + 1] : 0.0f;
    v2f b;
    b.x = vh[kk * 128 + d0 + ln];
    b.y = vh[(kk + 1) * 128 + d0 + ln];
    acc = wmma4(a, b, acc);
  }
  float* o = outh + (size_t)hb * 512 * 128;
#pragma unroll
  for (int r = 0; r < 8; ++r)
    o[(m0 + 8 * hi + r) * 128 + d0 + ln] = acc[r];
}

// ---------------------------------------------------------------------------
// h_all = concat(h_cache, h) along seq
// ---------------------------------------------------------------------------
__global__ __launch_bounds__(256)
void concat_kernel(float* __restrict__ dst, const float* __restrict__ h,
                   const float* __restrict__ hc) {
  int idx = blockIdx.x * 256 + threadIdx.x;     // 4*1024*1024 total
  int b = idx >> 20;
  int rem = idx & 1048575;
  int t = rem >> 10;
  int c = rem & 1023;
  float val = (t < 512) ? hc[((size_t)b * 512 + t) * 1024 + c]
                        : h[((size_t)b * 512 + (t - 512)) * 1024 + c];
  dst[idx] = val;
}

// ---------------------------------------------------------------------------
// out = LayerNorm(X + Y) * g + b      (one block per row of 1024)
// ---------------------------------------------------------------------------
__global__ __launch_bounds__(256)
void add_ln_kernel(const float* __restrict__ X, const float* __restrict__ Y,
                   const float* __restrict__ g, const float* __restrict__ b,
                   float* __restrict__ out) {
  const int tid = threadIdx.x;
  const float* x = X + (size_t)blockIdx.x * 1024;
  const float* y = Y + (size_t)blockIdx.x * 1024;
  float* o = out + (size_t)blockIdx.x * 1024;
  __shared__ float s1[256], s2[256];
  float vals[4], sum = 0.0f, sq = 0.0f;
#pragma unroll
  for (int i = 0; i < 4; ++i) {
    int c = tid + i * 256;
    float t = x[c] + y[c];
    vals[i] = t;
    sum += t;
    sq += t * t;
  }
  s1[tid] = sum;
  s2[tid] = sq;
  __syncthreads();
  for (int s = 128; s > 0; s >>= 1) {
    if (tid < s) { s1[tid] += s1[tid + s]; s2[tid] += s2[tid + s]; }
    __syncthreads();
  }
  float mean = s1[0] * (1.0f / 1024.0f);
  float var = s2[0] * (1.0f / 1024.0f) - mean * mean;
  float inv = rsqrtf(var + 1e-5f);
#pragma unroll
  for (int i = 0; i < 4; ++i) {
    int c = tid + i * 256;
    o[c] = (vals[i] - mean) * inv * g[c] + b[c];
  }
}

// ---------------------------------------------------------------------------
// Gate: logits = x @ gW + gb ; top-2 (jax tie-break: lowest index wins),
// softmax over the two; write top_idx (as float) to output tail; build
// per-expert token lists + counts with atomics.
// ---------------------------------------------------------------------------
__global__ __launch_bounds__(256)
void gate_kernel(const float* __restrict__ h1, const float* __restrict__ gW,
                 const float* __restrict__ gb, float* __restrict__ out_idx,
                 int* __restrict__ toks, float* __restrict__ wts,
                 int* __restrict__ cnt) {
  const int n = blockIdx.x;
  const int tid = threadIdx.x;
  const float* x = h1 + (size_t)n * 1024;
  float part[8] = {0, 0, 0, 0, 0, 0, 0, 0};
  for (int k = tid; k < 1024; k += 256) {
    float xv = x[k];
    const float* w = gW + (size_t)k * 8;
#pragma unroll
    for (int e = 0; e < 8; ++e) part[e] += xv * w[e];
  }
  __shared__ float sr[256];
  __shared__ float logit[8];
  for (int e = 0; e < 8; ++e) {
    sr[tid] = part[e];
    __syncthreads();
    for (int s = 128; s > 0; s >>= 1) {
      if (tid < s) sr[tid] += sr[tid + s];
      __syncthreads();
    }
    if (tid == 0) logit[e] = sr[0] + gb[e];
    __syncthreads();
  }
  if (tid == 0) {
    int i0 = 0;
    float v0 = logit[0];
    for (int e = 1; e < 8; ++e)
      if (logit[e] > v0) { v0 = logit[e]; i0 = e; }
    int i1 = -1;
    float v1 = -3.4e38f;
    for (int e = 0; e < 8; ++e) {
      if (e == i0) continue;
      if (logit[e] > v1) { v1 = logit[e]; i1 = e; }
    }
    float s0 = 1.0f / (1.0f + expf(v1 - v0));
    float s1 = 1.0f - s0;
    out_idx[n * 2 + 0] = (float)i0;
    out_idx[n * 2 + 1] = (float)i1;
    int p0 = atomicAdd(&cnt[i0], 1);
    toks[i0 * 2048 + p0] = n;
    wts[i0 * 2048 + p0] = s0;
    int p1 = atomicAdd(&cnt[i1], 1);
    toks[i1 * 2048 + p1] = n;
    wts[i1 * 2048 + p1] = s1;
  }
}

// ---------------------------------------------------------------------------
// MoE expert GEMM 1:  he[i,:2048] = relu(X[tok_i,:1024] @ W + b1)  for i<count
// ---------------------------------------------------------------------------
__global__ __launch_bounds__(128)
void moe_ffn1(const float* __restrict__ X, const float* __restrict__ W,
              const float* __restrict__ bias, const int* __restrict__ toks,
              const int* __restrict__ cnt, float* __restrict__ he) {
  const int count = *cnt;
  const int m_blk = blockIdx.y * 64;
  if (m_blk >= count) return;
  const int n_blk = blockIdx.x * 64;
  const int tid = threadIdx.x;
  const int wave = tid >> 5;
  const int lane = tid & 31;
  const int ln = lane & 15;
  const int hi = lane >> 4;

  __shared__ float As[64 * 36];
  __shared__ float Bs2[16 * 128];
  v8f acc[4];
#pragma unroll
  for (int t = 0; t < 4; ++t) acc[t] = v8zero();

  auto aload = [&](int r, int c4, int k0) -> float4 {
    int gm = m_blk + r;
    float4 v = make_float4(0.f, 0.f, 0.f, 0.f);
    if (gm < count) {
      int tok = toks[gm];
      v = *(const float4*)(X + (size_t)tok * 1024 + k0 + c4);
    }
    return v;
  };
  auto bload = [&](int r, int c4, int k0) -> float4 {
    return *(const float4*)(W + (size_t)(k0 + r) * 2048 + n_blk + c4);
  };
  gemm_tiles(1024, aload, bload, As, Bs2, acc, tid);

#pragma unroll
  for (int t = 0; t < 4; ++t) {
    int n = n_blk + t * 16 + ln;
#pragma unroll
    for (int r = 0; r < 8; ++r) {
      int gm = m_blk + wave * 16 + 8 * hi + r;
      if (gm < count)
        he[(size_t)gm * 2048 + n] = fmaxf(acc[t][r] + bias[n], 0.0f);
    }
  }
}

// ---------------------------------------------------------------------------
// MoE expert GEMM 2:  core[tok_i,:] += w_i * (he[i,:2048] @ W + b2)
// ---------------------------------------------------------------------------
__global__ __launch_bounds__(128)
void moe_ffn2(const float* __restrict__ he, const float* __restrict__ W,
              const float* __restrict__ bias, const int* __restrict__ toks,
              const float* __restrict__ wts, const int* __restrict__ cnt,
              float* __restrict__ core) {
  const int count = *cnt;
  const int m_blk = blockIdx.y * 64;
  if (m_blk >= count) return;
  const int n_blk = blockIdx.x * 64;
  const int tid = threadIdx.x;
  const int wave = tid >> 5;
  const int lane = tid & 31;
  const int ln = lane & 15;
  const int hi = lane >> 4;

  __shared__ float As[64 * 36];
  __shared__ float Bs2[16 * 128];
  v8f acc[4];
#pragma unroll
  for (int t = 0; t < 4; ++t) acc[t] = v8zero();

  auto aload = [&](int r, int c4, int k0) -> float4 {
    int gm = m_blk + r;
    float4 v = make_float4(0.f, 0.f, 0.f, 0.f);
    if (gm < count) v = *(const float4*)(he + (size_t)gm * 2048 + k0 + c4);
    return v;
  };
  auto bload = [&](int r, int c4, int k0) -> float4 {
    return *(const float4*)(W + (size_t)(k0 + r) * 1024 + n_blk + c4);
  };
  gemm_tiles(2048, aload, bload, As, Bs2, acc, tid);

#pragma unroll
  for (int t = 0; t < 4; ++t) {
    int n = n_blk + t * 16 + ln;
#pragma unroll
    for (int r = 0; r < 8; ++r) {
      int gm = m_blk + wave * 16 + 8 * hi + r;
      if (gm < count) {
        int tok = toks[gm];
        float w = wts[gm];
        atomicAdd(&core[(size_t)tok * 1024 + n], w * (acc[t][r] + bias[n]));
      }
    }
  }
}

// ---------------------------------------------------------------------------
extern "C" void kernel_launch(void* const* d_in, const int* in_sizes, int n_in,
                              void* d_out, int out_size, void* d_ws,
                              size_t ws_size, hipStream_t stream) {
  const float* h    = (const float*)d_in[0];
  const float* hc   = (const float*)d_in[1];
  const float* kpe  = (const float*)d_in[2];
  const float* Wq   = (const float*)d_in[3];
  const float* Wk   = (const float*)d_in[4];
  const float* Wv   = (const float*)d_in[5];
  const float* Wo   = (const float*)d_in[6];
  const float* ln1g = (const float*)d_in[7];
  const float* ln1b = (const float*)d_in[8];
  const float* gW   = (const float*)d_in[9];
  const float* gb   = (const float*)d_in[10];
  const float* W1   = (const float*)d_in[11];
  const float* b1   = (const float*)d_in[12];
  const float* W2   = (const float*)d_in[13];
  const float* b2   = (const float*)d_in[14];
  const float* ln2g = (const float*)d_in[15];
  const float* ln2b = (const float*)d_in[16];
  float* out = (float*)d_out;

  // Workspace layout (floats), with region reuse:
  float* fA = (float*)d_ws;        // 4,194,304 : h_all  -> he (per-expert)
  float* fB = fA + 4194304;        // 2,097,152 : q_heads -> core
  float* fC = fB + 2097152;        // 4,194,304 : k_heads -> attn_proj
  float* fD = fC + 4194304;        // 4,194,304 : v_heads -> h1
  float* fE = fD + 4194304;        // 8,388,608 : band    -> gate lists
  float* fF = fE + 8388608;        // 2,097,152 : attn head output
  int*   toks = (int*)fE;          // 8 * 2048 ints
  float* wts  = fE + 16384;        // 8 * 2048 floats
  int*   cnt  = (int*)(fE + 32768);// 8 ints
  float* outIdx = out + 4 * 512 * 1024;

  // 1. h_all = concat(h_cache, h)
  concat_kernel<<<16384, 256, 0, stream>>>(fA, h, hc);
  // 2-4. q/k/v projections, stored head-major [32,T,128]
  gemm_f32_wmma<<<dim3(16, 32, 1), 128, 0, stream>>>(
      h, Wq, fB, 2048, 1024, 1024, 1024, 1024, 0, 0, 0, 0, 512, 0);
  gemm_f32_wmma<<<dim3(16, 64, 1), 128, 0, stream>>>(
      fA, Wk, fC, 4096, 1024, 1024, 1024, 1024, 0, 0, 0, 0, 1024, 0);
  gemm_f32_wmma<<<dim3(16, 64, 1), 128, 0, stream>>>(
      fA, Wv, fD, 4096, 1024, 1024, 1024, 1024, 0, 0, 0, 0, 1024, 0);
  // 5. positional scores: band = q_h @ key_pe  (batched over 32 heads)
  gemm_f32_wmma<<<dim3(8, 8, 32), 128, 0, stream>>>(
      fB, kpe, fE, 512, 512, 128, 128, 512, 512,
      (long long)512 * 128, 0, (long long)512 * 512, 0, 0);
  // 6. content scores added into the band
  attn_band_cont<<<dim3(16, 32, 32), 128, 0, stream>>>(fB, fC, fE);
  // 7. scaled softmax over band rows
  softmax_band<<<16384, 256, 0, stream>>>(fE);
  // 8. out_h = skew(p) @ v_h
  attn_out_kernel<<<dim3(2, 32, 32), 128, 0, stream>>>(fE, fD, fF);
  // 9. attn_proj = merge_heads(out) @ Wo
  gemm_f32_wmma<<<dim3(16, 32, 1), 128, 0, stream>>>(
      fF, Wo, fC, 2048, 1024, 1024, 0, 1024, 1024, 0, 0, 0, 0, 512);
  // 10. h1 = LN(h + attn_proj)
  add_ln_kernel<<<2048, 256, 0, stream>>>(h, fC, ln1g, ln1b, fD);
  // 11. zero gate counts + MoE accumulator
  hipMemsetAsync(cnt, 0, 8 * sizeof(int), stream);
  hipMemsetAsync(fB, 0, (size_t)2097152 * sizeof(float), stream);
  // 12. gating (also writes top_idx to output tail)
  gate_kernel<<<2048, 256, 0, stream>>>(fD, gW, gb, outIdx, toks, wts, cnt);
  // 13. per-expert FFN (sparse: only routed tokens)
  for (int e = 0; e < 8; ++e) {
    moe_ffn1<<<dim3(32, 32), 128, 0, stream>>>(
        fD, W1 + (size_t)e * 1024 * 2048, b1 + (size_t)e * 2048,
        toks + e * 2048, cnt + e, fA);
    moe_ffn2<<<dim3(16, 32), 128, 0, stream>>>(
        fA, W2 + (size_t)e * 2048 * 1024, b2 + (size_t)e * 1024,
        toks + e * 2048, wts + e * 2048, cnt + e, fB);
  }
  // 14. output = LN(h1 + core)
  add_ln_kernel<<<2048, 256, 0, stream>>>(fD, fB, ln2g, ln2b, out);
}